// HeartbeatGNN_82806969467453
// MI455X (gfx1250) — compile-verified
//
#include <hip/hip_runtime.h>
#include <hip/hip_bf16.h>
#include <float.h>

typedef __attribute__((ext_vector_type(2))) float v2f;
typedef __attribute__((ext_vector_type(8))) float v8f;

#define NEG_SLOPE 0.2f
#define EPS_SM 1e-16f

static __device__ __forceinline__ float leaky(float v) {
  return v > 0.f ? v : NEG_SLOPE * v;
}

// -------------------------------------------------------------------------
// WMMA fp32 GEMM: C[M,N] = A[M,K] @ B[K,N] (+ optional bias, optional relu).
// One wave per 16x16 output tile, K stepped by 4 via V_WMMA_F32_16X16X4_F32.
// K is a template parameter so full steps have NO bounds guards (straight
// global_load_b64/b32 feeding wmma); the ragged tail (K=6) uses clamped
// addresses + value masks instead of exec-guarded loads.
// -------------------------------------------------------------------------
template <int K, bool RELU, bool HASBIAS>
__global__ void gemm_wmma_f32(const float* __restrict__ A, const float* __restrict__ B,
                              float* __restrict__ C, const float* __restrict__ bias,
                              int M, int N) {
  const int lane   = threadIdx.x & 31;
  const int laneLo = lane & 15;
  const int laneHi = lane >> 4;
  const int m0 = blockIdx.x << 4;
  const int n0 = blockIdx.y << 4;
  const int col = n0 + laneLo;
  v8f acc = {};
  const float* Arow = A + (size_t)(m0 + laneLo) * K;
  constexpr int KFULL = K & ~3;

#pragma unroll 4
  for (int kk = 0; kk < KFULL; kk += 4) {
    const int ka = kk + 2 * laneHi;
    v2f a, b;
    a.x = Arow[ka];                      // contiguous pair -> global_load_b64
    a.y = Arow[ka + 1];
    b.x = B[(size_t)ka * N + col];
    b.y = B[(size_t)(ka + 1) * N + col];
    acc = __builtin_amdgcn_wmma_f32_16x16x4_f32(false, a, false, b, (short)0, acc,
                                                false, false);
  }
  if constexpr (KFULL < K) {             // ragged tail (only K=6 hits this)
    const int ka  = KFULL + 2 * laneHi;
    const float mx = (ka     < K) ? 1.f : 0.f;
    const float my = (ka + 1 < K) ? 1.f : 0.f;
    const int  kax = (ka     < K) ? ka     : (K - 1);  // clamp: always in-bounds
    const int  kay = (ka + 1 < K) ? ka + 1 : (K - 1);
    v2f a, b;
    a.x = Arow[kax] * mx;
    a.y = Arow[kay] * my;
    b.x = B[(size_t)kax * N + col] * mx;
    b.y = B[(size_t)kay * N + col] * my;
    acc = __builtin_amdgcn_wmma_f32_16x16x4_f32(false, a, false, b, (short)0, acc,
                                                false, false);
  }

#pragma unroll
  for (int v = 0; v < 8; ++v) {
    float val = acc[v];
    if constexpr (HASBIAS) val += bias[col];   // only emit the add when real
    if constexpr (RELU)    val  = fmaxf(val, 0.f);
    C[(size_t)(m0 + v + 8 * laneHi) * N + col] = val;
  }
}

// ---- generic float fill ---------------------------------------------------
__global__ void fill_f32(float* __restrict__ p, float v, int n) {
  int i = blockIdx.x * blockDim.x + threadIdx.x;
  if (i < n) p[i] = v;
}

// ---- attention scores: a_src/a_dst = <h[n,hd,:], att[hd,:]> ---------------
__global__ void attn_scores(const float* __restrict__ h,
                            const float* __restrict__ att_s,
                            const float* __restrict__ att_d,
                            float* __restrict__ asrc, float* __restrict__ adst,
                            int N, int Hh, int Cc) {
  int i = blockIdx.x * blockDim.x + threadIdx.x;
  if (i >= N * Hh) return;
  int n = i / Hh, hd = i - n * Hh;
  const float* hp = h + ((size_t)n * Hh + hd) * Cc;
  const float* sp = att_s + hd * Cc;
  const float* dp = att_d + hd * Cc;
  float ss = 0.f, sd = 0.f;
  for (int c = 0; c < Cc; ++c) { float v = hp[c]; ss += v * sp[c]; sd += v * dp[c]; }
  asrc[i] = ss;
  adst[i] = sd;
}

static __device__ __forceinline__ void edge_ends(const int* __restrict__ ei, int E,
                                                 int e, int& s, int& d) {
  if (e < E) { s = ei[e]; d = ei[E + e]; }   // edge_index rows [0]=src, [1]=dst
  else       { s = d = e - E; }              // appended self-loops
}

// ---- pass 1: per-dst segment max of leakyrelu(a_src[s]+a_dst[d]) ----------
__global__ void edge_seg_max(const int* __restrict__ ei, int E, int Etot,
                             const float* __restrict__ asrc, const float* __restrict__ adst,
                             float* __restrict__ m, int Hh) {
  int i = blockIdx.x * blockDim.x + threadIdx.x;
  if (i >= Etot * Hh) return;
  int e = i / Hh, hd = i - e * Hh;
  int s, d; edge_ends(ei, E, e, s, d);
  float v = leaky(asrc[s * Hh + hd] + adst[d * Hh + hd]);
  float* addr = m + (size_t)d * Hh + hd;
  // float atomic-max via int max (v>=0) / uint min (v<0); m initialized -FLT_MAX
  if (v >= 0.f) atomicMax((int*)addr, __float_as_int(v));
  else          atomicMin((unsigned int*)addr, __float_as_uint(v));
}

// ---- pass 2: ex = exp(e - m[dst]); denom[dst] += ex -----------------------
__global__ void edge_exp_sum(const int* __restrict__ ei, int E, int Etot,
                             const float* __restrict__ asrc, const float* __restrict__ adst,
                             const float* __restrict__ m, float* __restrict__ exb,
                             float* __restrict__ den, int Hh) {
  int i = blockIdx.x * blockDim.x + threadIdx.x;
  if (i >= Etot * Hh) return;
  int e = i / Hh, hd = i - e * Hh;
  int s, d; edge_ends(ei, E, e, s, d);
  float v  = leaky(asrc[s * Hh + hd] + adst[d * Hh + hd]);
  float ex = __expf(v - m[(size_t)d * Hh + hd]);
  exb[i] = ex;
  atomicAdd(den + (size_t)d * Hh + hd, ex);
}

// ---- pass 2.5: exb -> alpha = exb / (den[dst]+eps)   (one divide / edge) --
__global__ void edge_alpha(const int* __restrict__ ei, int E, int Etot,
                           const float* __restrict__ den, float* __restrict__ exb,
                           int Hh) {
  int i = blockIdx.x * blockDim.x + threadIdx.x;
  if (i >= Etot * Hh) return;
  int e = i / Hh, hd = i - e * Hh;
  int s, d; edge_ends(ei, E, e, s, d);
  exb[i] = exb[i] / (den[(size_t)d * Hh + hd] + EPS_SM);
}

// ---- pass 3: out[dst] += alpha * h[src]  (thread per (edge, channel)) -----
__global__ void edge_aggregate(const int* __restrict__ ei, int E, int Etot,
                               const float* __restrict__ alpha,
                               const float* __restrict__ h, float* __restrict__ out,
                               int Hh, int Cc) {
  int i = blockIdx.x * blockDim.x + threadIdx.x;
  if (i >= Etot * Cc) return;
  int e = i / Cc, c = i - e * Cc;
  int s, d; edge_ends(ei, E, e, s, d);
  for (int hd = 0; hd < Hh; ++hd) {
    atomicAdd(out + ((size_t)d * Hh + hd) * Cc + c,
              h[((size_t)s * Hh + hd) * Cc + c] * alpha[(size_t)e * Hh + hd]);
  }
}

// ---- x = relu(x + bias[col]) ---------------------------------------------
__global__ void bias_relu(float* __restrict__ x, const float* __restrict__ b,
                          int N, int F) {
  int i = blockIdx.x * blockDim.x + threadIdx.x;
  if (i >= N * F) return;
  x[i] = fmaxf(x[i] + b[i % F], 0.f);
}

// ---- global mean pool: sum + count, then divide ---------------------------
__global__ void pool_sum(const float* __restrict__ h, const int* __restrict__ batch,
                         float* __restrict__ pooled, float* __restrict__ cnt,
                         int N, int F) {
  int i = blockIdx.x * blockDim.x + threadIdx.x;
  if (i >= N * F) return;
  int n = i / F, c = i - n * F;
  int g = batch[n];
  atomicAdd(pooled + (size_t)g * F + c, h[i]);
  if (c == 0) atomicAdd(cnt + g, 1.f);
}

__global__ void pool_div(float* __restrict__ pooled, const float* __restrict__ cnt,
                         int G, int F) {
  int i = blockIdx.x * blockDim.x + threadIdx.x;
  if (i >= G * F) return;
  pooled[i] /= fmaxf(cnt[i / F], 1.f);
}

static inline int cdiv(int a, int b) { return (a + b - 1) / b; }

extern "C" void kernel_launch(void* const* d_in, const int* in_sizes, int n_in,
                              void* d_out, int out_size, void* d_ws, size_t ws_size,
                              hipStream_t stream) {
  const float* x     = (const float*)d_in[0];
  const int*   ei    = (const int*)  d_in[1];   // [2,E] (src row, dst row)
  const int*   batch = (const int*)  d_in[2];
  const float* W1    = (const float*)d_in[3];   // [6,128]
  const float* as1   = (const float*)d_in[4];   // [4,32]
  const float* ad1   = (const float*)d_in[5];
  const float* b1    = (const float*)d_in[6];   // [128]
  const float* W2    = (const float*)d_in[7];   // [128,64]
  const float* as2   = (const float*)d_in[8];   // [1,64]
  const float* ad2   = (const float*)d_in[9];
  const float* b2    = (const float*)d_in[10];  // [64]
  const float* Wout  = (const float*)d_in[11];  // [64,32]
  const float* bout  = (const float*)d_in[12];  // [32]
  float*       out   = (float*)d_out;           // [512,32]

  const int N    = in_sizes[0] / 6;             // 50000 (multiple of 16)
  const int E    = in_sizes[1] / 2;             // 800000
  const int Etot = E + N;                       // + self-loops
  const int G    = 512;
  const int H1 = 4, C1 = 32, F1 = 128;          // layer 1: 4 heads x 32
  const int F2 = 64;                            // layer 2: 1 head x 64

  // ---- workspace layout (floats), with layer-2 aliasing layer-1 regions ---
  float* ws    = (float*)d_ws;
  float* h1    = ws;                            // [N,128]
  float* out1  = h1    + (size_t)N * F1;        // [N,128]
  float* asrc1 = out1  + (size_t)N * F1;        // [N,4]
  float* adst1 = asrc1 + (size_t)N * H1;        // [N,4]
  float* m1    = adst1 + (size_t)N * H1;        // [N,4]
  float* den1  = m1    + (size_t)N * H1;        // [N,4]
  float* ex1   = den1  + (size_t)N * H1;        // [Etot,4]
  // layer-2 aliases (layer-1 temps dead by then):
  float* h2    = h1;                            // [N,64]  (fits in h1 region)
  float* out2  = h1 + (size_t)N * F2;           // [N,64]  (second half of h1 region)
  float* asrc2 = asrc1;                         // [N]
  float* adst2 = adst1;                         // [N]
  float* m2    = m1;                            // [N]
  float* den2  = den1;                          // [N]
  float* ex2   = ex1;                           // [Etot]
  float* pooled= ex1 + (size_t)Etot;            // [G,64] (inside ex1 region, used post-edges)
  float* cnt   = pooled + (size_t)G * F2;       // [G]

  const int TB = 256;

  // ============================ layer 1 ====================================
  // h1 = x @ W1   (K=6, tail handled with clamped/masked loads)
  gemm_wmma_f32<6, false, false>
      <<<dim3(N / 16, F1 / 16), 32, 0, stream>>>(x, W1, h1, nullptr, N, F1);
  attn_scores<<<cdiv(N * H1, TB), TB, 0, stream>>>(h1, as1, ad1, asrc1, adst1, N, H1, C1);

  fill_f32<<<cdiv(N * H1, TB), TB, 0, stream>>>(m1,   -FLT_MAX, N * H1);
  fill_f32<<<cdiv(N * H1, TB), TB, 0, stream>>>(den1, 0.f,      N * H1);
  fill_f32<<<cdiv(N * F1, TB), TB, 0, stream>>>(out1, 0.f,      N * F1);

  edge_seg_max <<<cdiv(Etot * H1, TB), TB, 0, stream>>>(ei, E, Etot, asrc1, adst1, m1, H1);
  edge_exp_sum <<<cdiv(Etot * H1, TB), TB, 0, stream>>>(ei, E, Etot, asrc1, adst1, m1, ex1, den1, H1);
  edge_alpha   <<<cdiv(Etot * H1, TB), TB, 0, stream>>>(ei, E, Etot, den1, ex1, H1);
  edge_aggregate<<<cdiv(Etot * C1, TB), TB, 0, stream>>>(ei, E, Etot, ex1, h1, out1, H1, C1);
  bias_relu<<<cdiv(N * F1, TB), TB, 0, stream>>>(out1, b1, N, F1);

  // ============================ layer 2 ====================================
  // h2 = out1 @ W2
  gemm_wmma_f32<128, false, false>
      <<<dim3(N / 16, F2 / 16), 32, 0, stream>>>(out1, W2, h2, nullptr, N, F2);
  attn_scores<<<cdiv(N, TB), TB, 0, stream>>>(h2, as2, ad2, asrc2, adst2, N, 1, F2);

  fill_f32<<<cdiv(N, TB), TB, 0, stream>>>(m2,   -FLT_MAX, N);
  fill_f32<<<cdiv(N, TB), TB, 0, stream>>>(den2, 0.f,      N);
  fill_f32<<<cdiv(N * F2, TB), TB, 0, stream>>>(out2, 0.f, N * F2);

  edge_seg_max <<<cdiv(Etot, TB), TB, 0, stream>>>(ei, E, Etot, asrc2, adst2, m2, 1);
  edge_exp_sum <<<cdiv(Etot, TB), TB, 0, stream>>>(ei, E, Etot, asrc2, adst2, m2, ex2, den2, 1);
  edge_alpha   <<<cdiv(Etot, TB), TB, 0, stream>>>(ei, E, Etot, den2, ex2, 1);
  edge_aggregate<<<cdiv(Etot * F2, TB), TB, 0, stream>>>(ei, E, Etot, ex2, h2, out2, 1, F2);
  bias_relu<<<cdiv(N * F2, TB), TB, 0, stream>>>(out2, b2, N, F2);

  // ======================= pool + final linear =============================
  fill_f32<<<cdiv(G * F2, TB), TB, 0, stream>>>(pooled, 0.f, G * F2);
  fill_f32<<<cdiv(G, TB), TB, 0, stream>>>(cnt, 0.f, G);
  pool_sum<<<cdiv(N * F2, TB), TB, 0, stream>>>(out2, batch, pooled, cnt, N, F2);
  pool_div<<<cdiv(G * F2, TB), TB, 0, stream>>>(pooled, cnt, G, F2);

  // out = pooled @ Wout + bout
  gemm_wmma_f32<64, false, true>
      <<<dim3(G / 16, 32 / 16), 32, 0, stream>>>(pooled, Wout, out, bout, G, 32);
}